// TemporalMultipyPool3D_53721450939020
// MI455X (gfx1250) — compile-verified
//
#include <hip/hip_runtime.h>
#include <math.h>

// x: (8, 512, 16, 28, 28) fp32.  out: (B*C, 7) fp32 = [n1 | n2_0 n2_1 | n4_0..3]
#define T_DIM     16
#define W_DIM     28
#define H_DIM     28
#define PLANE     (W_DIM * H_DIM)       // 784 floats per frame
#define VOL       (T_DIM * PLANE)       // 12544 floats per (b,c)
#define VOL_BYTES (VOL * 4)             // 50176 bytes
#define CHUNKS    (VOL_BYTES / 16)      // 3136 x 16B async transfers
#define NTHREADS  256

// One 16B async global->LDS copy; OFF is applied by HW to BOTH the LDS and
// global addresses (CDNA5 ISA), so a single base pair serves
// all iterations.  NT: data is streamed exactly once, bypass temporal caching.
template <int OFF>
__device__ __forceinline__ void async_cp16(unsigned lds, unsigned long long g) {
    asm volatile("global_load_async_to_lds_b128 %0, %1, off offset:%2 th:TH_LOAD_NT"
                 :: "v"(lds), "v"(g), "n"(OFF) : "memory");
}

__global__ __launch_bounds__(NTHREADS)
void tmp3d_kernel(const float* __restrict__ x, float* __restrict__ out)
{
    __shared__ __align__(16) float smem[VOL];   // staged (b,c) volume
    __shared__ float tileMax[64];               // fine maxes: [tb][wy][sx]

    const int bc  = blockIdx.x;        // b*512 + c
    const int tid = threadIdx.x;

    // ---- Stage whole volume into LDS: CDNA5 async copies, immediate offsets ----
    {
        const unsigned lbase =
            (unsigned)(uintptr_t)(&smem[0]) + (unsigned)(tid * 16);
        const unsigned long long gbase =
            (unsigned long long)(uintptr_t)(x + (size_t)bc * VOL) + (unsigned)(tid * 16);

        // chunks: addr = base + tid*16 + i*4096  (i*256 chunks per step)
        async_cp16<0 * 4096>(lbase, gbase);
        async_cp16<1 * 4096>(lbase, gbase);
        async_cp16<2 * 4096>(lbase, gbase);
        async_cp16<3 * 4096>(lbase, gbase);
        async_cp16<4 * 4096>(lbase, gbase);
        async_cp16<5 * 4096>(lbase, gbase);
        async_cp16<6 * 4096>(lbase, gbase);
        async_cp16<7 * 4096>(lbase, gbase);
        async_cp16<8 * 4096>(lbase, gbase);
        async_cp16<9 * 4096>(lbase, gbase);
        async_cp16<10 * 4096>(lbase, gbase);
        async_cp16<11 * 4096>(lbase, gbase);
        if (tid < 64)                              // tail: 3136 - 12*256 = 64
            async_cp16<12 * 4096>(lbase, gbase);

        asm volatile("s_wait_asynccnt 0" ::: "memory");
    }
    __syncthreads();

    // ---- Fine max: 64 tiles of 4 frames x 7x7; 4 threads per tile ----
    const int thr  = tid & 3;          // which frame of the tile's 4
    const int tile = tid >> 2;         // 0..63
    const int tb   = tile >> 4;        // temporal block 0..3
    const int wy   = (tile >> 2) & 3;  // spatial row tile 0..3
    const int sx   = tile & 3;         // spatial col tile 0..3

    const float* base = &smem[(4 * tb + thr) * PLANE + (7 * wy) * W_DIM + 7 * sx];
    float m = -INFINITY;
    #pragma unroll
    for (int yo = 0; yo < 7; ++yo) {
        const float* row = base + yo * W_DIM;
        #pragma unroll
        for (int xo = 0; xo < 7; ++xo)
            m = fmaxf(m, row[xo]);
    }
    // combine the 4 per-frame partials (lanes 4k..4k+3) -> tile max
    m = fmaxf(m, __shfl_xor(m, 1));
    m = fmaxf(m, __shfl_xor(m, 2));
    if (thr == 0) tileMax[tile] = m;
    __syncthreads();

    // ---- Hierarchy: 64 fine maxes -> 7 outputs (trivial, lane 0) ----
    if (tid == 0) {
        float* o = out + (size_t)bc * 7;

        // n = 1 : max over everything
        float o1 = -INFINITY;
        #pragma unroll
        for (int i = 0; i < 64; ++i) o1 = fmaxf(o1, tileMax[i]);
        o[0] = o1;

        // n = 2 : per temporal half, mean of 4 quadrant maxes (each = max of 8 fine)
        #pragma unroll
        for (int t2 = 0; t2 < 2; ++t2) {
            float s = 0.0f;
            #pragma unroll
            for (int qy = 0; qy < 2; ++qy) {
                #pragma unroll
                for (int qx = 0; qx < 2; ++qx) {
                    float q = -INFINITY;
                    #pragma unroll
                    for (int dt = 0; dt < 2; ++dt)
                        #pragma unroll
                        for (int dy = 0; dy < 2; ++dy)
                            #pragma unroll
                            for (int dx = 0; dx < 2; ++dx)
                                q = fmaxf(q, tileMax[(2*t2 + dt)*16 + (2*qy + dy)*4 + (2*qx + dx)]);
                    s += q;
                }
            }
            o[1 + t2] = s * 0.25f;
        }

        // n = 4 : per temporal quarter, mean of the 16 fine maxes
        #pragma unroll
        for (int t4 = 0; t4 < 4; ++t4) {
            float s = 0.0f;
            #pragma unroll
            for (int sp = 0; sp < 16; ++sp) s += tileMax[t4 * 16 + sp];
            o[3 + t4] = s * (1.0f / 16.0f);
        }
    }
}

extern "C" void kernel_launch(void* const* d_in, const int* in_sizes, int n_in,
                              void* d_out, int out_size, void* d_ws, size_t ws_size,
                              hipStream_t stream) {
    (void)n_in; (void)out_size; (void)d_ws; (void)ws_size;
    const float* x = (const float*)d_in[0];
    float* out = (float*)d_out;
    const int n_bc = in_sizes[0] / VOL;   // 8*512 = 4096 blocks
    tmp3d_kernel<<<n_bc, NTHREADS, 0, stream>>>(x, out);
}